// GCN_82506321756776
// MI455X (gfx1250) — compile-verified
//
#include <hip/hip_runtime.h>
#include <hip/hip_bf16.h>

typedef __attribute__((ext_vector_type(2))) float v2f;
typedef __attribute__((ext_vector_type(8))) float v8f;

#define NN 100000   // 100000 % 16 == 0 -> GEMM M-tiles cover exactly, no guards
#define EE 3200000
#define FIN 20
#define FHID 20
#define FOUT 256

// ---------------------------------------------------------------------------
// Stage 1: scatter x[src] into Z[dst][0..19] (layer-1 mean-agg numerator),
// and accumulate degree (shared by both layers).
// One thread per (edge, feature).
// ---------------------------------------------------------------------------
__global__ __launch_bounds__(256) void scatter1_kernel(
    const float* __restrict__ x, const long long* __restrict__ ei,
    float* __restrict__ Z, float* __restrict__ deg) {
  long long t = (long long)blockIdx.x * blockDim.x + threadIdx.x;
  if (t >= (long long)EE * FIN) return;
  int e = (int)(t / FIN);
  int f = (int)(t % FIN);
  long long src = ei[e];
  long long dst = ei[(long long)EE + e];
  atomicAdd(&Z[dst * 40 + f], x[src * FIN + f]);
  if (f == 0) atomicAdd(&deg[dst], 1.0f);
}

// ---------------------------------------------------------------------------
// Stage 2: per-node dense part of layer 1:
//   h = relu( (Z[i][0..19]*inv_deg) @ W_l1 + b_l1 + x[i] @ W_r1 )
// writes h into Z[i][20..39] and re-zeroes Z[i][0..19] so Z becomes the
// layer-2 scatter accumulator. Weights cached in LDS. All row traffic is
// float4 (rows of x are 80 B / rows of Z are 160 B, both 16 B aligned).
// ---------------------------------------------------------------------------
__global__ __launch_bounds__(256) void apply1_kernel(
    const float* __restrict__ x, const float* __restrict__ deg,
    const float* __restrict__ Wl1, const float* __restrict__ bl1,
    const float* __restrict__ Wr1, float* __restrict__ Z) {
  __shared__ float sWl[FIN * FHID];
  __shared__ float sWr[FIN * FHID];
  __shared__ float sb[FHID];
  for (int i = threadIdx.x; i < FIN * FHID; i += blockDim.x) {
    sWl[i] = Wl1[i];
    sWr[i] = Wr1[i];
  }
  if (threadIdx.x < FHID) sb[threadIdx.x] = bl1[threadIdx.x];
  __syncthreads();

  int i = blockIdx.x * blockDim.x + threadIdx.x;
  if (i >= NN) return;

  float inv = 1.0f / fmaxf(deg[i], 1.0f);

  const float4* zr = (const float4*)(Z + (long long)i * 40);
  const float4* xr = (const float4*)(x + (long long)i * FIN);
  float a[FIN], xv[FIN];
#pragma unroll
  for (int q = 0; q < 5; ++q) {
    float4 za = zr[q];
    float4 xa = xr[q];
    a[q * 4 + 0] = za.x * inv; a[q * 4 + 1] = za.y * inv;
    a[q * 4 + 2] = za.z * inv; a[q * 4 + 3] = za.w * inv;
    xv[q * 4 + 0] = xa.x; xv[q * 4 + 1] = xa.y;
    xv[q * 4 + 2] = xa.z; xv[q * 4 + 3] = xa.w;
  }

  // reset cols 0..19 -> accumulator for layer-2 scatter
  float4* zw = (float4*)(Z + (long long)i * 40);
  float4 z4 = make_float4(0.f, 0.f, 0.f, 0.f);
#pragma unroll
  for (int q = 0; q < 5; ++q) zw[q] = z4;

  float h[FHID];
#pragma unroll
  for (int j = 0; j < FHID; ++j) {
    float s = sb[j];
#pragma unroll
    for (int k = 0; k < FIN; ++k)
      s += a[k] * sWl[k * FHID + j] + xv[k] * sWr[k * FHID + j];
    h[j] = fmaxf(s, 0.0f);
  }
#pragma unroll
  for (int q = 0; q < 5; ++q)
    zw[5 + q] = make_float4(h[q * 4 + 0], h[q * 4 + 1], h[q * 4 + 2], h[q * 4 + 3]);
}

// ---------------------------------------------------------------------------
// Stage 3: scatter h[src] (= Z[src][20..39]) into Z[dst][0..19].
// ---------------------------------------------------------------------------
__global__ __launch_bounds__(256) void scatter2_kernel(
    const long long* __restrict__ ei, float* __restrict__ Z) {
  long long t = (long long)blockIdx.x * blockDim.x + threadIdx.x;
  if (t >= (long long)EE * FHID) return;
  int e = (int)(t / FHID);
  int f = (int)(t % FHID);
  long long src = ei[e];
  long long dst = ei[(long long)EE + e];
  atomicAdd(&Z[dst * 40 + f], Z[src * 40 + 20 + f]);
}

// ---------------------------------------------------------------------------
// Stage 4: WMMA GEMM: out = [agg2*inv_deg , h] @ [W_l2 ; W_r2] + b_l2
//   A = Z (inv_deg folded into k<20 loads), [100000 x 40]
//   B = concat(W_l2, W_r2) row-major [40 x 256], staged in LDS
// Block: 16 M-rows x 256 N-cols; 8 waves x 2 N-tiles; K=40 -> 10x
// V_WMMA_F32_16X16X4_F32 per tile.
// A layout (ISA 16x4 f32): lane L -> M=L%16, Kpair=(L/16)*2.
// C/D layout: lane L -> N=L%16, VGPR v -> M = v + 8*(L/16).
// NN % 16 == 0 and grid == NN/16, so no M guards anywhere (EXEC stays full,
// stores are straight-line with immediate offsets).
// ---------------------------------------------------------------------------
__global__ __launch_bounds__(256) void gemm2_kernel(
    const float* __restrict__ Z, const float* __restrict__ deg,
    const float* __restrict__ Wl2, const float* __restrict__ Wr2,
    const float* __restrict__ bl2, float* __restrict__ out) {
  __shared__ float sW[40 * FOUT];   // 40 KB
  __shared__ float sB[FOUT];
  for (int i = threadIdx.x; i < FHID * FOUT; i += 256) {
    sW[i] = Wl2[i];                 // rows 0..19
    sW[FHID * FOUT + i] = Wr2[i];   // rows 20..39
  }
  if (threadIdx.x < FOUT) sB[threadIdx.x] = bl2[threadIdx.x];
  __syncthreads();

  const int wave = threadIdx.x >> 5;  // 0..7
  const int lane = threadIdx.x & 31;
  const int lmod = lane & 15;
  const int lhi  = lane >> 4;         // 0 or 1
  const int m_base = blockIdx.x * 16;

  const int mrow = m_base + lmod;     // always < NN
  const float inv = 1.0f / fmaxf(deg[mrow], 1.0f);
  const int koff = lhi * 2;           // 0 or 2

  const float* arow = Z + (long long)mrow * 40 + koff;
  v2f A[10];
#pragma unroll
  for (int kk = 0; kk < 10; ++kk) {
    // k = kk*4 + koff is even; pair (k, k+1) is 8-byte aligned -> b64 load
    v2f av = *(const v2f*)(arow + kk * 4);
    if (kk * 4 + koff < 20) { av.x *= inv; av.y *= inv; }
    A[kk] = av;
  }

#pragma unroll
  for (int nt = 0; nt < 2; ++nt) {
    int n = (wave * 2 + nt) * 16 + lmod;
    v8f acc = {};
#pragma unroll
    for (int kk = 0; kk < 10; ++kk) {
      int k = kk * 4 + koff;
      v2f b;
      b.x = sW[k * FOUT + n];
      b.y = sW[(k + 1) * FOUT + n];
      acc = __builtin_amdgcn_wmma_f32_16x16x4_f32(
          /*neg_a=*/false, A[kk], /*neg_b=*/false, b,
          /*c_mod=*/(short)0, acc, /*reuse_a=*/false, /*reuse_b=*/false);
    }
    float bias = sB[n];
    float* op = out + (long long)(m_base + 8 * lhi) * FOUT + n;
#pragma unroll
    for (int v = 0; v < 8; ++v)
      op[(long long)v * FOUT] = acc[v] + bias;   // immediate-offset stores
  }
}

// ---------------------------------------------------------------------------
// Launch
// Workspace layout (floats): deg[N] | Z[N*40]   -> 16.4 MB, zeroed per call.
// ---------------------------------------------------------------------------
extern "C" void kernel_launch(void* const* d_in, const int* in_sizes, int n_in,
                              void* d_out, int out_size, void* d_ws, size_t ws_size,
                              hipStream_t stream) {
  const float* x        = (const float*)d_in[0];
  const long long* ei   = (const long long*)d_in[1];
  const float* Wl1      = (const float*)d_in[2];
  const float* bl1      = (const float*)d_in[3];
  const float* Wr1      = (const float*)d_in[4];
  const float* Wl2      = (const float*)d_in[5];
  const float* bl2      = (const float*)d_in[6];
  const float* Wr2      = (const float*)d_in[7];
  float* out            = (float*)d_out;

  float* deg = (float*)d_ws;
  float* Z   = deg + NN;

  size_t zero_bytes = (size_t)NN * 41 * sizeof(float);
  hipMemsetAsync(d_ws, 0, zero_bytes, stream);

  {
    long long total = (long long)EE * FIN;
    int grid = (int)((total + 255) / 256);
    scatter1_kernel<<<grid, 256, 0, stream>>>(x, ei, Z, deg);
  }
  {
    int grid = (NN + 255) / 256;
    apply1_kernel<<<grid, 256, 0, stream>>>(x, deg, Wl1, bl1, Wr1, Z);
  }
  {
    long long total = (long long)EE * FHID;
    int grid = (int)((total + 255) / 256);
    scatter2_kernel<<<grid, 256, 0, stream>>>(ei, Z);
  }
  {
    int grid = NN / 16;  // 6250, exact cover
    gemm2_kernel<<<grid, 256, 0, stream>>>(Z, deg, Wl2, Wr2, bl2, out);
  }
}